// NURBSSurfaces_40587440947487
// MI455X (gfx1250) — compile-verified
//
#include <hip/hip_runtime.h>
#include <stdint.h>

typedef float v2f __attribute__((ext_vector_type(2)));

namespace {
constexpr int S_DIM = 16, F_DIM = 4, P_DIM = 16384;
constexpr int NUC = 64, NVC = 64, DEG = 3;
constexpr int BLOCK = 256;            // 8 waves of 32 on CDNA5
constexpr int PPT = 8;                // points per thread
constexpr int PTS_PER_BLOCK = BLOCK * PPT;             // 2048
constexpr int BLOCKS_PER_SURF = P_DIM / PTS_PER_BLOCK; // 8
constexpr int N_SURF = S_DIM * F_DIM;                  // 64
constexpr int TOTAL_PTS = N_SURF * P_DIM;              // 1,048,576
constexpr int CP_FLOATS = NUC * NVC * 3;               // 12288 floats
constexpr int CP_BYTES = CP_FLOATS * 4;                // 49152 B of LDS
constexpr int COPY_ITERS = CP_BYTES / (BLOCK * 16);    // 12 x b128 per lane
constexpr float KNOT_SCALE = 1.0f / 61.0f;             // n_ctrl - degree intervals
}

// scalar v_rcp_f32 + one Newton-Raphson step
__device__ __forceinline__ float fast_rcp(float x) {
  float r = __builtin_amdgcn_rcpf(x);
  return r * (2.0f - x * r);
}

// packed: 2x v_rcp_f32 + packed Newton-Raphson (pk_fma/pk_mul)
__device__ __forceinline__ v2f fast_rcp2(v2f x) {
  v2f r = (v2f){__builtin_amdgcn_rcpf(x.x), __builtin_amdgcn_rcpf(x.y)};
  return r * (2.0f - x * r);
}

// v_rsq_f32 + one Newton-Raphson step
__device__ __forceinline__ float fast_rsqrt(float x) {
  float r = __builtin_amdgcn_rsqf(x);
  return r * (1.5f - 0.5f * x * r * r);
}

// Dual-direction cubic Cox-de-Boor basis + 1st derivative, packed as (u,v)
// pairs so the whole recurrence runs on v_pk_* 2-wide f32 math.
// All divisors are knot differences; the derivative divisors are exactly the
// level-3 knot differences, so their reciprocals are reused.
__device__ __forceinline__ void cubic_basis2(v2f t, v2f bh,
                                             v2f N[4], v2f dN[4]) {
  // knot(span+k) = clamp(bh + k*h, 0, 1); v_med3 is lane-scalar -> 2 per knot.
  auto knotR = [&](int k) -> v2f {
    v2f x = bh + (float)k * KNOT_SCALE;
    return (v2f){__builtin_amdgcn_fmed3f(x.x, 0.0f, 1.0f),
                 __builtin_amdgcn_fmed3f(x.y, 0.0f, 1.0f)};
  };
  v2f left[DEG + 1], right[DEG + 1];
  v2f ndu[DEG + 1][DEG + 1];
  v2f inv3[DEG];                  // 1 / ndu[3][r], r = 0..2
  ndu[0][0] = (v2f){1.0f, 1.0f};
#pragma unroll
  for (int j = 1; j <= DEG; ++j) {
    left[j]  = t - knotR(1 - j);
    right[j] = knotR(j) - t;
    v2f saved = (v2f){0.0f, 0.0f};
#pragma unroll
    for (int r = 0; r < j; ++r) {
      const v2f den = right[r + 1] + left[j - r];   // knot difference
      const v2f inv = fast_rcp2(den);
      if (j == DEG) inv3[r] = inv;
      const v2f temp = ndu[r][j - 1] * inv;
      ndu[r][j] = saved + right[r + 1] * temp;
      saved = left[j - r] * temp;
    }
    ndu[j][j] = saved;
  }
#pragma unroll
  for (int r = 0; r <= DEG; ++r) N[r] = ndu[r][DEG];
  // q[r] = p * ndu[r][p-1] / ndu[p][r]; dN[r] = q[r-1] - q[r] (q[-1]=q[3]=0)
  v2f q[DEG];
#pragma unroll
  for (int r = 0; r < DEG; ++r) q[r] = (float)DEG * ndu[r][DEG - 1] * inv3[r];
  dN[0] = -q[0];
  dN[1] = q[0] - q[1];
  dN[2] = q[1] - q[2];
  dN[3] = q[2];
}

__global__ void __launch_bounds__(BLOCK)
nurbs_surface_eval(const float* __restrict__ ev,   // [S,F,P,2]
                   const float* __restrict__ cps,  // [S,F,64,64,3]
                   float* __restrict__ out) {      // points then normals
  __shared__ float cp_lds[CP_FLOATS];              // 48 KB

  const int b      = blockIdx.x;
  const int surf   = b / BLOCKS_PER_SURF;
  const int chunk  = b % BLOCKS_PER_SURF;
  const int basePt = surf * P_DIM + chunk * PTS_PER_BLOCK;

  // ---- Stage this surface's 48KB control grid into LDS (async DMA path) ----
  const void* surfCp = (const void*)(cps + (size_t)surf * CP_FLOATS);
  const unsigned ldsBase = (unsigned)(uintptr_t)(&cp_lds[0]); // low 32b = LDS off
#if defined(__HIP_DEVICE_COMPILE__)
#pragma unroll
  for (int it = 0; it < COPY_ITERS; ++it) {
    unsigned byteOff = (unsigned)((it * BLOCK + threadIdx.x) * 16);
    unsigned ldsAddr = ldsBase + byteOff;
    // GLOBAL_LOAD_ASYNC_TO_LDS_B128: vdst = LDS addr, vaddr = 32b offset, saddr = base
    asm volatile("global_load_async_to_lds_b128 %0, %1, %2 offset:0"
                 :
                 : "v"(ldsAddr), "v"(byteOff), "s"(surfCp)
                 : "memory");
  }
  asm volatile("s_wait_asynccnt 0x0" ::: "memory");
#else
  for (int i = threadIdx.x; i < CP_FLOATS; i += BLOCK)
    cp_lds[i] = ((const float*)surfCp)[i];
#endif
  __syncthreads();

#pragma unroll 1
  for (int r = 0; r < PPT; ++r) {
    const int p = basePt + r * BLOCK + (int)threadIdx.x;
    const float2 uv = reinterpret_cast<const float2*>(ev)[p];
    const float u = uv.x, v = uv.y;

    // u,v >= 0, so integer truncation == floor.
    int spanU = (int)(u * (float)(NUC - DEG)) + DEG;
    int spanV = (int)(v * (float)(NVC - DEG)) + DEG;
    spanU = spanU > NUC - 1 ? NUC - 1 : spanU;
    spanV = spanV > NVC - 1 ? NVC - 1 : spanV;

    // Packed dual-direction basis: lane .x = u direction, lane .y = v direction.
    const v2f tuv = (v2f){u, v};
    const v2f bh  = (v2f){(float)(spanU - DEG) * KNOT_SCALE,
                          (float)(spanV - DEG) * KNOT_SCALE};
    v2f Nb[4], dNb[4];
    cubic_basis2(tuv, bh, Nb, dNb);

    const int iu0 = spanU - DEG;
    const int iv0 = spanV - DEG;

    // Homogeneous w channel: all cp weights are 1, so
    // SKL[k,l].w = (sum_i Bu[k,i]) * (sum_j Bv[l,j]).  Packed sums.
    const v2f sumN = Nb[0] + Nb[1] + Nb[2] + Nb[3];    // (su0, sv0)
    const v2f sumD = dNb[0] + dNb[1] + dNb[2] + dNb[3];// (su1, sv1)
    const float w00h = sumN.x * sumN.y;   // SKL[0,0].w  (numerically ~1)
    const float w10h = sumD.x * sumN.y;   // SKL[1,0].w  (numerically ~0)
    const float w01h = sumN.x * sumD.y;   // SKL[0,1].w  (numerically ~0)

    // Packed (basis, derivative) v-pairs -> v_pk_fma_f32 2-wide math.
    v2f bvv[DEG + 1];
#pragma unroll
    for (int j = 0; j <= DEG; ++j) bvv[j] = (v2f){Nb[j].y, dNb[j].y};

    v2f A0x = {0.f, 0.f}, A0y = {0.f, 0.f}, A0z = {0.f, 0.f}; // (a00, a01)
    float a10x = 0.f, a10y = 0.f, a10z = 0.f;
#pragma unroll
    for (int i = 0; i <= DEG; ++i) {
      const int rowOff = (iu0 + i) * NVC + iv0;
      v2f rx = {0.f, 0.f}, ry = {0.f, 0.f}, rz = {0.f, 0.f}; // (sum Nv*cp, sum dNv*cp)
#pragma unroll
      for (int j = 0; j <= DEG; ++j) {
        const float* c = &cp_lds[(rowOff + j) * 3];
        const float x = c[0], y = c[1], z = c[2];
        const v2f bvj = bvv[j];
        rx += bvj * x;
        ry += bvj * y;
        rz += bvj * z;
      }
      const float bu0 = Nb[i].x, bu1 = dNb[i].x;
      A0x += bu0 * rx; A0y += bu0 * ry; A0z += bu0 * rz;
      a10x += bu1 * rx.x; a10y += bu1 * ry.x; a10z += bu1 * rz.x;
    }
    const float a00x = A0x.x, a00y = A0y.x, a00z = A0z.x;
    const float a01x = A0x.y, a01y = A0y.y, a01z = A0z.y;

    // Rational projection with one shared reciprocal of w00.
    const float invw = fast_rcp(w00h);
    const float px = a00x * invw, py = a00y * invw, pz = a00z * invw;
    const float sux = (a10x - w10h * px) * invw;
    const float suy = (a10y - w10h * py) * invw;
    const float suz = (a10z - w10h * pz) * invw;
    const float svx = (a01x - w01h * px) * invw;
    const float svy = (a01y - w01h * py) * invw;
    const float svz = (a01z - w01h * pz) * invw;

    float nx = suy * svz - suz * svy;
    float ny = suz * svx - sux * svz;
    float nz = sux * svy - suy * svx;
    const float rn = fast_rsqrt(nx * nx + ny * ny + nz * nz);
    nx *= rn; ny *= rn; nz *= rn;

    float4 outP = make_float4(px, py, pz, 1.0f);
    float4 outN = make_float4(nx, ny, nz, 0.0f);
    reinterpret_cast<float4*>(out)[p] = outP;
    reinterpret_cast<float4*>(out)[TOTAL_PTS + p] = outN;
  }
}

extern "C" void kernel_launch(void* const* d_in, const int* in_sizes, int n_in,
                              void* d_out, int out_size, void* d_ws, size_t ws_size,
                              hipStream_t stream) {
  (void)in_sizes; (void)n_in; (void)out_size; (void)d_ws; (void)ws_size;
  const float* ev  = (const float*)d_in[0];   // evaluation_points [16,4,16384,2]
  const float* cps = (const float*)d_in[1];   // control_points   [16,4,64,64,3]
  // d_in[2], d_in[3] are degree_u/degree_v == 3, hardcoded.
  float* out = (float*)d_out;                 // points [S,F,P,4] then normals [S,F,P,4]

  dim3 grid(N_SURF * BLOCKS_PER_SURF);        // 512 blocks
  dim3 block(BLOCK);
  nurbs_surface_eval<<<grid, block, 0, stream>>>(ev, cps, out);
}